// ExpertMLP_MPNet_17961553232376
// MI455X (gfx1250) — compile-verified
//
#include <hip/hip_runtime.h>
#include <math.h>

typedef __attribute__((ext_vector_type(2))) float v2f;
typedef __attribute__((ext_vector_type(8))) float v8f;

#define BATCH 512      // B
#define DIM   512      // DIN = DH = DOUT
#define NEXP  8        // experts
#define KTOT  (NEXP * DIM)   // 4096
#define KC    32       // K chunk staged in LDS (divides DIM -> single expert per chunk)
#define BM    64       // block tile M
#define BN    64       // block tile N

// One blended-expert layer: out[b,n] = act( sum_e coef[b,e] * (x @ w[e])[b,n] + sum_e coef[b,e]*bias[e,n] )
// Implemented as GEMM: A[b, e*DIM+i] = coef[b,e]*x[b,i]  (formed on the fly), Bmat = w viewed as [KTOT, DIM].
template <bool APPLY_ELU>
__global__ __launch_bounds__(128) void moe_layer_kernel(
    const float* __restrict__ in,    // [BATCH, DIM]
    const float* __restrict__ coef,  // [BATCH, NEXP]
    const float* __restrict__ w,     // [NEXP, DIM, DIM] == [KTOT, DIM]
    const float* __restrict__ bias,  // [NEXP, DIM]
    float* __restrict__ out)         // [BATCH, DIM]
{
    __shared__ float As[BM][KC + 2];   // row stride 34 floats (136B, 8B-aligned rows)
    __shared__ float Bs[KC][BN + 2];   // row stride 66 floats
    __shared__ float Cf[BM][NEXP];     // coef tile for this block's rows

    const int t   = threadIdx.x;
    const int wv  = t >> 5;            // wave id 0..3
    const int L   = t & 31;            // lane
    const int lhi = L >> 4;            // 0/1 half-wave
    const int llo = L & 15;
    const int wm  = wv >> 1;           // 2x2 wave grid
    const int wn  = wv & 1;
    const int m0  = blockIdx.y * BM;
    const int n0  = blockIdx.x * BN;

    // ---- load coef tile: 64 rows x 8 = 128 float4, one per thread ----
    {
        const float4* cg = (const float4*)(coef + (size_t)(m0 + (t >> 1)) * NEXP) + (t & 1);
        float4 c4 = *cg;
        float* dst = &Cf[t >> 1][(t & 1) * 4];
        dst[0] = c4.x; dst[1] = c4.y; dst[2] = c4.z; dst[3] = c4.w;
    }

    v8f acc[2][2] = {};   // 2x2 WMMA 16x16 f32 tiles per wave

    const int arow  = t >> 1;        // A loader: row 0..63, half-row per thread
    const int acol4 = (t & 1) * 4;   // starting float4 within the 8-float4 row chunk
    const int brow  = t >> 2;        // B loader: k-row 0..31
    const int bcol4 = (t & 3) * 4;   // starting float4 within 16-float4 row

    for (int k0 = 0; k0 < KTOT; k0 += KC) {
        const int e  = k0 >> 9;          // expert for this chunk (KC divides DIM)
        const int ko = k0 & (DIM - 1);   // offset inside x row

        __syncthreads();  // previous tile fully consumed; also publishes Cf on iter 0

        // A tile: scale x by per-row expert coefficient while staging
        {
            const float cs = Cf[arow][e];
            const float4* ag = (const float4*)(in + (size_t)(m0 + arow) * DIM + ko) + acol4;
#pragma unroll
            for (int r = 0; r < 4; ++r) {
                float4 v = ag[r];
                float* dst = &As[arow][(acol4 + r) * 4];
                dst[0] = v.x * cs; dst[1] = v.y * cs; dst[2] = v.z * cs; dst[3] = v.w * cs;
            }
        }
        // B tile: straight copy of w[k0+brow][n0 .. n0+63]
        {
            const float4* bg = (const float4*)(w + (size_t)(k0 + brow) * DIM + n0) + bcol4;
#pragma unroll
            for (int r = 0; r < 4; ++r) {
                float4 v = bg[r];
                float* dst = &Bs[brow][(bcol4 + r) * 4];
                dst[0] = v.x; dst[1] = v.y; dst[2] = v.z; dst[3] = v.w;
            }
        }

        __syncthreads();

        // ---- WMMA inner loop: f32 16x16x4, ISA register layouts ----
#pragma unroll
        for (int ks = 0; ks < KC; ks += 4) {
            const int kr = ks + 2 * lhi;
            // A frag: lane L -> A[M=L%16][K=2*(L/16)+j], j=0,1 : one 8B LDS read
            v2f a0 = *(const v2f*)&As[wm * 32 +      llo][kr];
            v2f a1 = *(const v2f*)&As[wm * 32 + 16 + llo][kr];
            // B frag: lane L -> B[K=2*(L/16)+j][N=L%16]
            v2f b0, b1;
            b0.x = Bs[kr][wn * 32 + llo];        b0.y = Bs[kr + 1][wn * 32 + llo];
            b1.x = Bs[kr][wn * 32 + 16 + llo];   b1.y = Bs[kr + 1][wn * 32 + 16 + llo];

            acc[0][0] = __builtin_amdgcn_wmma_f32_16x16x4_f32(false, a0, false, b0, (short)0, acc[0][0], false, false);
            acc[0][1] = __builtin_amdgcn_wmma_f32_16x16x4_f32(false, a0, false, b1, (short)0, acc[0][1], false, false);
            acc[1][0] = __builtin_amdgcn_wmma_f32_16x16x4_f32(false, a1, false, b0, (short)0, acc[1][0], false, false);
            acc[1][1] = __builtin_amdgcn_wmma_f32_16x16x4_f32(false, a1, false, b1, (short)0, acc[1][1], false, false);
        }
    }

    // ---- epilogue: blended bias + optional ELU; C layout lane L vgpr j -> (M=j+8*(L/16), N=L%16) ----
#pragma unroll
    for (int mt = 0; mt < 2; ++mt) {
#pragma unroll
        for (int nt = 0; nt < 2; ++nt) {
            const int col = n0 + wn * 32 + nt * 16 + llo;
            float be[NEXP];
#pragma unroll
            for (int ee = 0; ee < NEXP; ++ee) be[ee] = bias[ee * DIM + col];
            const int rbase = wm * 32 + mt * 16 + 8 * lhi;
#pragma unroll
            for (int j = 0; j < 8; ++j) {
                const int rl = rbase + j;
                float bb = 0.0f;
#pragma unroll
                for (int ee = 0; ee < NEXP; ++ee) bb += Cf[rl][ee] * be[ee];
                float v = acc[mt][nt][j] + bb;
                if (APPLY_ELU) v = (v > 0.0f) ? v : (__expf(v) - 1.0f);
                out[(size_t)(m0 + rl) * DIM + col] = v;
            }
        }
    }
}

extern "C" void kernel_launch(void* const* d_in, const int* in_sizes, int n_in,
                              void* d_out, int out_size, void* d_ws, size_t ws_size,
                              hipStream_t stream) {
    const float* X    = (const float*)d_in[0];
    const float* coef = (const float*)d_in[1];
    const float* w1   = (const float*)d_in[2];
    const float* b1   = (const float*)d_in[3];
    const float* w2   = (const float*)d_in[4];
    const float* b2   = (const float*)d_in[5];
    const float* w3   = (const float*)d_in[6];
    const float* b3   = (const float*)d_in[7];
    float* out = (float*)d_out;

    float* h1 = (float*)d_ws;                    // [512,512] = 1 MB
    float* h2 = h1 + (size_t)BATCH * DIM;        // [512,512] = 1 MB

    dim3 grid(DIM / BN, BATCH / BM);             // 8 x 8 = 64 blocks
    dim3 block(128);                             // 4 wave32s

    moe_layer_kernel<true ><<<grid, block, 0, stream>>>(X,  coef, w1, b1, h1);
    moe_layer_kernel<true ><<<grid, block, 0, stream>>>(h1, coef, w2, b2, h2);
    moe_layer_kernel<false><<<grid, block, 0, stream>>>(h2, coef, w3, b3, out);
}